// SpliceGraph_53334903882135
// MI455X (gfx1250) — compile-verified
//
#include <hip/hip_runtime.h>
#include <math.h>

typedef __attribute__((ext_vector_type(2))) float v2f;
typedef __attribute__((ext_vector_type(8))) float v8f;

#define HCH 128              // hidden / in channels
#define EPSBN 1e-5f

// ---------------------------------------------------------------------------
// Degree / normalization
// ---------------------------------------------------------------------------
__global__ __launch_bounds__(256) void deg_init_kernel(float* deg, int n) {
  int i = blockIdx.x * blockDim.x + threadIdx.x;
  if (i < n) deg[i] = 1.0f;                       // self loop
}

__global__ __launch_bounds__(256) void deg_edges_kernel(const int* __restrict__ dst,
                                                        float* deg, int e) {
  int i = blockIdx.x * blockDim.x + threadIdx.x;
  if (i < e) atomicAdd(&deg[dst[i]], 1.0f);
}

__global__ __launch_bounds__(256) void dinv_kernel(float* deg, int n) {
  int i = blockIdx.x * blockDim.x + threadIdx.x;
  if (i < n) deg[i] = rsqrtf(deg[i]);             // in-place deg -> deg^-1/2
}

// ---------------------------------------------------------------------------
// WMMA fp32 GEMM: Y[n x 128] = X[n x 128] @ W[128 x 128] (+ bias)
// block = 256 threads = 8 waves; block covers 16 rows x 128 cols.
// Each wave owns one 16x16 N-tile; A strip staged in LDS and shared by 8 waves.
// Requires n % 16 == 0 (N = 50000 -> 3125 blocks exactly).
// ---------------------------------------------------------------------------
__global__ __launch_bounds__(256) void gemm128_wmma_kernel(const float* __restrict__ X,
                                                           const float* __restrict__ W,
                                                           const float* __restrict__ bias,
                                                           float* __restrict__ Y,
                                                           int hasBias) {
  __shared__ float sX[16 * HCH];                  // 8 KB A strip
  const int tid  = threadIdx.x;
  const int wave = tid >> 5;                      // 0..7 -> N tile
  const int lane = tid & 31;
  const int rowBase = blockIdx.x * 16;

  // cooperative load of 16x128 A strip (512 float4, 2 per thread)
  {
    const float4* s = (const float4*)(X + (size_t)rowBase * HCH);
    float4* d = (float4*)sX;
    d[tid]       = s[tid];
    d[tid + 256] = s[tid + 256];
  }
  __syncthreads();

  const int half  = lane >> 4;                    // 0: lanes 0-15, 1: lanes 16-31
  const int arow  = lane & 15;                    // local A row (M)
  const int col   = wave * 16 + (lane & 15);      // output column (N)
  const int khalf = half * 2;                     // K sub-offset 0 or 2

  v8f c = {};
#pragma unroll
  for (int k0 = 0; k0 < HCH; k0 += 4) {
    // A 16x4 frag: VGPR0 = K(khalf), VGPR1 = K(khalf+1)   (ISA 7.12.2)
    v2f a = *(const v2f*)(&sX[arow * HCH + k0 + khalf]);
    // B 4x16 frag: VGPR0 = row K(khalf), VGPR1 = row K(khalf+1), N striped on lanes
    v2f b;
    b.x = W[(size_t)(k0 + khalf + 0) * HCH + col];
    b.y = W[(size_t)(k0 + khalf + 1) * HCH + col];
    c = __builtin_amdgcn_wmma_f32_16x16x4_f32(false, a, false, b, (short)0, c,
                                              false, false);
  }

  const float bv = hasBias ? bias[col] : 0.0f;
  const int r0 = rowBase + half * 8;              // C/D: VGPR v = rows v / v+8
#pragma unroll
  for (int v = 0; v < 8; ++v)
    Y[(size_t)(r0 + v) * HCH + col] = c[v] + bv;
}

// ---------------------------------------------------------------------------
// acc[i] = t[i] * dinv[i]^2   (self-loop contribution; also initializes acc)
// ---------------------------------------------------------------------------
__global__ __launch_bounds__(256) void selfloop_kernel(const float* __restrict__ t,
                                                       const float* __restrict__ dinv,
                                                       float* __restrict__ acc,
                                                       size_t nh) {
  size_t i = (size_t)blockIdx.x * blockDim.x + threadIdx.x;
  if (i < nh) {
    float dv = dinv[i >> 7];
    acc[i] = t[i] * dv * dv;
  }
}

// ---------------------------------------------------------------------------
// Edge scatter: acc[dst] += t[src] * dinv[src]*dinv[dst]
// 32 threads per edge, 4 channels each (float4 gather, scalar f32 atomics).
// ---------------------------------------------------------------------------
__global__ __launch_bounds__(256) void scatter_kernel(const float* __restrict__ t,
                                                      const float* __restrict__ dinv,
                                                      const int* __restrict__ src,
                                                      const int* __restrict__ dst,
                                                      float* __restrict__ acc,
                                                      int e) {
  int tid = blockIdx.x * blockDim.x + threadIdx.x;
  int ed  = tid >> 5;
  if (ed >= e) return;
  int c4 = (tid & 31) << 2;
  int s = src[ed];
  int d = dst[ed];
  float nrm = dinv[s] * dinv[d];
  // hint the next edge's source row toward the caches
  if (ed + 1 < e) __builtin_prefetch(&t[(size_t)src[ed + 1] * HCH + c4], 0, 0);
  float4 v = *(const float4*)(t + (size_t)s * HCH + c4);
  float* o = acc + (size_t)d * HCH + c4;
  atomicAdd(o + 0, v.x * nrm);
  atomicAdd(o + 1, v.y * nrm);
  atomicAdd(o + 2, v.z * nrm);
  atomicAdd(o + 3, v.w * nrm);
}

// ---------------------------------------------------------------------------
// z = tanh(acc + conv_b[c])
// ---------------------------------------------------------------------------
__global__ __launch_bounds__(256) void tanh_bias_kernel(const float* __restrict__ acc,
                                                        const float* __restrict__ b,
                                                        float* __restrict__ z,
                                                        size_t nh) {
  size_t i = (size_t)blockIdx.x * blockDim.x + threadIdx.x;
  if (i < nh) z[i] = tanhf(acc[i] + b[i & (HCH - 1)]);
}

// ---------------------------------------------------------------------------
// out = relu((1-sigmoid(gpre)) * hbase + sigmoid(gpre) * z)
// ---------------------------------------------------------------------------
__global__ __launch_bounds__(256) void combine_relu_kernel(const float* __restrict__ gpre,
                                                           const float* __restrict__ z,
                                                           const float* __restrict__ hbase,
                                                           float* __restrict__ out,
                                                           size_t nh) {
  size_t i = (size_t)blockIdx.x * blockDim.x + threadIdx.x;
  if (i < nh) {
    float g = 1.0f / (1.0f + expf(-gpre[i]));
    float h = (1.0f - g) * hbase[i] + g * z[i];
    out[i] = fmaxf(h, 0.0f);
  }
}

// ---------------------------------------------------------------------------
// BatchNorm: per-channel sums, finalize, apply
// ---------------------------------------------------------------------------
__global__ __launch_bounds__(256) void zero_kernel(float* p, int n) {
  int i = blockIdx.x * blockDim.x + threadIdx.x;
  if (i < n) p[i] = 0.0f;
}

__global__ __launch_bounds__(128) void bn_stats_kernel(const float* __restrict__ x,
                                                       float* __restrict__ sums,
                                                       int rowsPerBlock, int nRows) {
  int c = threadIdx.x;                            // channel
  int r0 = blockIdx.x * rowsPerBlock;
  int r1 = r0 + rowsPerBlock; if (r1 > nRows) r1 = nRows;
  float s = 0.0f, ss = 0.0f;
  for (int r = r0; r < r1; ++r) {
    float v = x[(size_t)r * HCH + c];
    s += v; ss += v * v;
  }
  atomicAdd(&sums[c], s);
  atomicAdd(&sums[HCH + c], ss);
}

__global__ __launch_bounds__(128) void bn_finalize_kernel(const float* __restrict__ sums,
                                                          const float* __restrict__ w,
                                                          const float* __restrict__ b,
                                                          float* __restrict__ scaleshift,
                                                          int nRows) {
  int c = threadIdx.x;
  float invn = 1.0f / (float)nRows;
  float mean = sums[c] * invn;
  float var  = sums[HCH + c] * invn - mean * mean;   // biased variance
  float a = w[c] * rsqrtf(var + EPSBN);
  scaleshift[c]       = a;
  scaleshift[HCH + c] = b[c] - mean * a;
}

__global__ __launch_bounds__(256) void bn_apply_kernel(const float* __restrict__ x,
                                                       const float* __restrict__ scaleshift,
                                                       float* __restrict__ y,
                                                       size_t nh) {
  size_t i = (size_t)blockIdx.x * blockDim.x + threadIdx.x;
  if (i < nh) {
    int c = (int)(i & (HCH - 1));
    y[i] = x[i] * scaleshift[c] + scaleshift[HCH + c];
  }
}

// ---------------------------------------------------------------------------
// Orchestration
// ---------------------------------------------------------------------------
extern "C" void kernel_launch(void* const* d_in, const int* in_sizes, int n_in,
                              void* d_out, int out_size, void* d_ws, size_t ws_size,
                              hipStream_t stream) {
  (void)n_in; (void)out_size; (void)ws_size;

  const float* x       = (const float*)d_in[0];
  const int*   ei      = (const int*)d_in[1];
  const float* lin_w   = (const float*)d_in[2];
  const float* lin_b   = (const float*)d_in[3];
  const float* conv1_w = (const float*)d_in[4];
  const float* conv1_b = (const float*)d_in[5];
  const float* gate1_w = (const float*)d_in[6];
  const float* gate1_b = (const float*)d_in[7];
  const float* bn1_w   = (const float*)d_in[8];
  const float* bn1_b   = (const float*)d_in[9];
  const float* conv2_w = (const float*)d_in[10];
  const float* conv2_b = (const float*)d_in[11];
  const float* gate2_w = (const float*)d_in[12];
  const float* gate2_b = (const float*)d_in[13];
  const float* bn2_w   = (const float*)d_in[14];
  const float* bn2_b   = (const float*)d_in[15];

  const int N = in_sizes[0] / HCH;        // 50000
  const int E = in_sizes[1] / 2;          // 1600000
  const size_t NH = (size_t)N * HCH;
  const int* srcv = ei;
  const int* dstv = ei + E;

  // workspace layout (floats)
  float* wsf   = (float*)d_ws;
  float* dinv  = wsf;                     // N  (deg, then deg^-1/2 in place)
  float* bufA  = dinv + N;                // N*H : t / gate pre-act
  float* bufB  = bufA + NH;               // N*H : acc / hlin / h2
  float* bufC  = bufB + NH;               // N*H : z
  float* bufD  = bufC + NH;               // N*H : h (block1 output)
  float* bnsum = bufD + NH;               // 256 : per-channel sum, sumsq
  float* bnss  = bnsum + 2 * HCH;         // 256 : scale, shift

  const int TB = 256;
  const int gN   = (N + TB - 1) / TB;
  const int gE   = (E + TB - 1) / TB;
  const int gNH  = (int)((NH + TB - 1) / TB);
  const int gSc  = (E * 32 + TB - 1) / TB;         // 32 threads / edge
  const int gMM  = N / 16;                          // N % 16 == 0
  const int rowsPerBlk = 250;                       // 200 blocks * 250 = 50000
  const int gBN  = (N + rowsPerBlk - 1) / rowsPerBlk;

  // ---- degree / normalization (shared by both blocks) ----
  deg_init_kernel<<<gN, TB, 0, stream>>>(dinv, N);
  deg_edges_kernel<<<gE, TB, 0, stream>>>(dstv, dinv, E);
  dinv_kernel<<<gN, TB, 0, stream>>>(dinv, N);

  // ================= block 1 =================
  gemm128_wmma_kernel<<<gMM, TB, 0, stream>>>(x, conv1_w, nullptr, bufA, 0);
  selfloop_kernel<<<gNH, TB, 0, stream>>>(bufA, dinv, bufB, NH);
  scatter_kernel<<<gSc, TB, 0, stream>>>(bufA, dinv, srcv, dstv, bufB, E);
  tanh_bias_kernel<<<gNH, TB, 0, stream>>>(bufB, conv1_b, bufC, NH);
  gemm128_wmma_kernel<<<gMM, TB, 0, stream>>>(bufC, gate1_w, gate1_b, bufA, 1);
  gemm128_wmma_kernel<<<gMM, TB, 0, stream>>>(x, lin_w, lin_b, bufB, 1);
  combine_relu_kernel<<<gNH, TB, 0, stream>>>(bufA, bufC, bufB, bufD, NH);
  zero_kernel<<<1, TB, 0, stream>>>(bnsum, 2 * HCH);
  bn_stats_kernel<<<gBN, HCH, 0, stream>>>(bufD, bnsum, rowsPerBlk, N);
  bn_finalize_kernel<<<1, HCH, 0, stream>>>(bnsum, bn1_w, bn1_b, bnss, N);
  bn_apply_kernel<<<gNH, TB, 0, stream>>>(bufD, bnss, bufD, NH);

  // ================= block 2 =================
  gemm128_wmma_kernel<<<gMM, TB, 0, stream>>>(bufD, conv2_w, nullptr, bufA, 0);
  selfloop_kernel<<<gNH, TB, 0, stream>>>(bufA, dinv, bufB, NH);
  scatter_kernel<<<gSc, TB, 0, stream>>>(bufA, dinv, srcv, dstv, bufB, E);
  tanh_bias_kernel<<<gNH, TB, 0, stream>>>(bufB, conv2_b, bufC, NH);
  gemm128_wmma_kernel<<<gMM, TB, 0, stream>>>(bufC, gate2_w, gate2_b, bufA, 1);
  combine_relu_kernel<<<gNH, TB, 0, stream>>>(bufA, bufC, bufD, bufB, NH);
  zero_kernel<<<1, TB, 0, stream>>>(bnsum, 2 * HCH);
  bn_stats_kernel<<<gBN, HCH, 0, stream>>>(bufB, bnsum, rowsPerBlk, N);
  bn_finalize_kernel<<<1, HCH, 0, stream>>>(bnsum, bn2_w, bn2_b, bnss, N);
  bn_apply_kernel<<<gNH, TB, 0, stream>>>(bufB, bnss, (float*)d_out, NH);
}